// OurLinear_90039694394026
// MI455X (gfx1250) — compile-verified
//
#include <hip/hip_runtime.h>

typedef float v2f __attribute__((ext_vector_type(2)));
typedef float v8f __attribute__((ext_vector_type(8)));
typedef int   v4i __attribute__((ext_vector_type(4)));

#define D 768
#define BS 64
#define SEQ 256
#define RANK 8
#define NV 24        // n = d / 32
#define QV 32        // VIEW_R
#define KC 32        // K chunk staged in LDS
#define XS_STRIDE 36 // pad: 36*L mod 64 -> 16 distinct banks, float4-aligned
#define NCHUNK (D / KC)

#if __has_builtin(__builtin_amdgcn_global_load_async_to_lds_b128) && \
    __has_builtin(__builtin_amdgcn_s_wait_asynccnt)
#define USE_ASYNC 1
#else
#define USE_ASYNC 0
#endif

// ---------------- W_eff = W + 2 * B @ A  (fold LoRA into base weight) ----------------
__global__ __launch_bounds__(256) void weff_kernel(
    const float* __restrict__ W, const float* __restrict__ A,
    const float* __restrict__ B, float* __restrict__ Weff) {
  int idx = blockIdx.x * 256 + threadIdx.x;
  if (idx >= D * D) return;
  int o = idx / D, i2 = idx % D;
  float acc = W[idx];
#pragma unroll
  for (int r = 0; r < RANK; r++)
    acc += 2.0f * B[o * RANK + r] * A[r * D + i2];
  Weff[idx] = acc;
}

// ---------------- wq[b, o] = sum_k h[b,k] * Wlin[o,k]  (per-branch, 64x768) ----------------
__global__ __launch_bounds__(256) void wq_kernel(
    const float* __restrict__ h, const float* __restrict__ Wlin,
    float* __restrict__ wq) {
  __shared__ float hs[D];
  const int b = blockIdx.y;
  const int o = blockIdx.x * 256 + threadIdx.x;
  for (int i = threadIdx.x; i < D; i += 256) hs[i] = h[b * D + i];
  __syncthreads();
  const float* wr = Wlin + o * D;
  float acc = 0.0f;
#pragma unroll 4
  for (int k = 0; k < D; k += 4) {
    float4 wv = *(const float4*)&wr[k];
    acc += wv.x * hs[k] + wv.y * hs[k + 1] + wv.z * hs[k + 2] + wv.w * hs[k + 3];
  }
  wq[b * D + o] = acc;
}

// ---------------- out[b,s,:] = bias + kron_user + kron_item (overwrites d_out) ----------------
__global__ __launch_bounds__(256) void kron_bias_kernel(
    const float* __restrict__ x, const float* __restrict__ bias,
    const float* __restrict__ VU, const float* __restrict__ VI,
    const float* __restrict__ wqu, const float* __restrict__ wqi,
    float* __restrict__ out) {
  __shared__ float xr[D];
  __shared__ float vu[NV * QV], vi[NV * QV];
  __shared__ float wu[D], wi[D];
  __shared__ float tu[QV][QV + 1], ti[QV][QV + 1];
  const int s = blockIdx.x, b = blockIdx.y;
  const int tid = threadIdx.x;
  const long rowbase = (long)(b * SEQ + s) * D;
  for (int i = tid; i < D; i += 256) {
    xr[i] = x[rowbase + i];
    wu[i] = wqu[b * D + i];
    wi[i] = wqi[b * D + i];
  }
  for (int i = tid; i < NV * QV; i += 256) {
    vu[i] = VU[i];
    vi[i] = VI[i];
  }
  __syncthreads();
  // t[q][m] = sum_n x[q*24+n] * V[n][m]   for both branches
  for (int p = tid; p < QV * QV; p += 256) {
    int q = p >> 5, m = p & 31;
    float su = 0.0f, si = 0.0f;
#pragma unroll
    for (int n = 0; n < NV; n++) {
      float xv = xr[q * NV + n];
      su += xv * vu[n * QV + m];
      si += xv * vi[n * QV + m];
    }
    tu[q][m] = su;
    ti[q][m] = si;
  }
  __syncthreads();
  // out[o = l*32+m] = bias[o] + sum_q t[q][m] * wq[q*24+l]
  for (int o = tid; o < D; o += 256) {
    int l = o >> 5, m = o & 31;
    float acc = bias[o];
#pragma unroll
    for (int q = 0; q < QV; q++) {
      acc += tu[q][m] * wu[q * NV + l];
      acc += ti[q][m] * wi[q * NV + l];
    }
    out[rowbase + o] = acc;
  }
}

// ---------------- main GEMM: out += x @ W_effᵀ   (WMMA f32 16x16x4) ----------------
// Workgroup = 256 threads (8 waves), tile 128 rows x 64 cols.
// Waves arranged 4 (row) x 2 (col); each wave owns a 32x32 register tile
// (four v8f accumulators -> 4 independent WMMA chains per K-step).

__device__ __forceinline__ void compute_chunk(
    const float (*Xb)[XS_STRIDE], const float (*Wb)[XS_STRIDE],
    int rA0, int cB0, int khalf, v8f& c00, v8f& c01, v8f& c10, v8f& c11) {
#pragma unroll
  for (int kk = 0; kk < KC; kk += 4) {
    v2f a0 = *(const v2f*)&Xb[rA0][kk + khalf];
    v2f a1 = *(const v2f*)&Xb[rA0 + 16][kk + khalf];
    v2f b0 = *(const v2f*)&Wb[cB0][kk + khalf];
    v2f b1 = *(const v2f*)&Wb[cB0 + 16][kk + khalf];
    c00 = __builtin_amdgcn_wmma_f32_16x16x4_f32(false, a0, false, b0, (short)0, c00, false, false);
    c01 = __builtin_amdgcn_wmma_f32_16x16x4_f32(false, a0, false, b1, (short)0, c01, false, false);
    c10 = __builtin_amdgcn_wmma_f32_16x16x4_f32(false, a1, false, b0, (short)0, c10, false, false);
    c11 = __builtin_amdgcn_wmma_f32_16x16x4_f32(false, a1, false, b1, (short)0, c11, false, false);
  }
}

__global__ __launch_bounds__(256) void gemm_wmma_kernel(
    const float* __restrict__ X, const float* __restrict__ Weff,
    float* __restrict__ out) {
#if USE_ASYNC
  __shared__ float Xs[2][128][XS_STRIDE];
  __shared__ float Ws[2][64][XS_STRIDE];
#else
  __shared__ float Xs[1][128][XS_STRIDE];
  __shared__ float Ws[1][64][XS_STRIDE];
#endif
  const int tid = threadIdx.x;
  const int row0 = blockIdx.y * 128;
  const int col0 = blockIdx.x * 64;
  const int lane = tid & 31;
  const int w = tid >> 5;
  const int rA0 = (w & 3) * 32 + (lane & 15);   // A rows for this wave's tile
  const int cB0 = (w >> 2) * 32 + (lane & 15);  // B cols for this wave's tile
  const int khalf = (lane >> 4) * 2;            // lanes 0-15: K{k,k+1}; 16-31: K{k+2,k+3}
  // cooperative staging coordinates: 8 threads per row, float4 per thread
  const int lr = tid >> 3;       // 0..31
  const int lc = (tid & 7) * 4;  // 0..28 step 4

  const float* Xg = X + (long)(row0 + lr) * D + lc;     // + k0 per chunk
  const float* Wg = Weff + (long)(col0 + lr) * D + lc;

  v8f c00 = {0.f, 0.f, 0.f, 0.f, 0.f, 0.f, 0.f, 0.f};
  v8f c01 = c00, c10 = c00, c11 = c00;

#if USE_ASYNC
  // -------- double-buffered async global->LDS pipeline (ASYNCcnt-tracked) --------
#define STAGE_ASYNC(bufi, k0)                                              \
  {                                                                        \
    _Pragma("unroll") for (int u = 0; u < 4; u++)                          \
        __builtin_amdgcn_global_load_async_to_lds_b128(                    \
            (v4i*)(Xg + (k0) + (long)(u * 32) * D),                        \
            (v4i*)&Xs[bufi][lr + u * 32][lc], 0, 0);                       \
    _Pragma("unroll") for (int u = 0; u < 2; u++)                          \
        __builtin_amdgcn_global_load_async_to_lds_b128(                    \
            (v4i*)(Wg + (k0) + (long)(u * 32) * D),                        \
            (v4i*)&Ws[bufi][lr + u * 32][lc], 0, 0);                       \
  }
  STAGE_ASYNC(0, 0)
  for (int chunk = 0; chunk < NCHUNK; ++chunk) {
    const int cur = chunk & 1;
    __builtin_amdgcn_s_wait_asynccnt(0);  // this chunk's LDS data landed
    __syncthreads();                      // visible to all waves; prev compute done
    if (chunk + 1 < NCHUNK) {
      if (cur == 0) STAGE_ASYNC(1, (chunk + 1) * KC)
      else          STAGE_ASYNC(0, (chunk + 1) * KC)
    }
    compute_chunk(Xs[cur], Ws[cur], rA0, cB0, khalf, c00, c01, c10, c11);
  }
#undef STAGE_ASYNC
#else
  // -------- synchronous fallback: global->VGPR->LDS --------
  for (int chunk = 0; chunk < NCHUNK; ++chunk) {
    const int k0 = chunk * KC;
    float4 xa0 = *(const float4*)(Xg + k0);
    float4 xa1 = *(const float4*)(Xg + k0 + 32L * D);
    float4 xa2 = *(const float4*)(Xg + k0 + 64L * D);
    float4 xa3 = *(const float4*)(Xg + k0 + 96L * D);
    float4 wv0 = *(const float4*)(Wg + k0);
    float4 wv1 = *(const float4*)(Wg + k0 + 32L * D);
    __syncthreads();
    *(float4*)&Xs[0][lr][lc] = xa0;
    *(float4*)&Xs[0][lr + 32][lc] = xa1;
    *(float4*)&Xs[0][lr + 64][lc] = xa2;
    *(float4*)&Xs[0][lr + 96][lc] = xa3;
    *(float4*)&Ws[0][lr][lc] = wv0;
    *(float4*)&Ws[0][lr + 32][lc] = wv1;
    __syncthreads();
    compute_chunk(Xs[0], Ws[0], rA0, cB0, khalf, c00, c01, c10, c11);
  }
#endif

  // epilogue: D layout — VGPR v: M = v + 8*(lane>=16), N = lane&15
  const int orow = row0 + (w & 3) * 32 + 8 * (lane >> 4);
  const int ocol = col0 + (w >> 2) * 32 + (lane & 15);
#pragma unroll
  for (int v = 0; v < 8; v++) {
    out[(long)(orow + v) * D + ocol] += c00[v];
    out[(long)(orow + v) * D + ocol + 16] += c01[v];
    out[(long)(orow + 16 + v) * D + ocol] += c10[v];
    out[(long)(orow + 16 + v) * D + ocol + 16] += c11[v];
  }
}

extern "C" void kernel_launch(void* const* d_in, const int* in_sizes, int n_in,
                              void* d_out, int out_size, void* d_ws, size_t ws_size,
                              hipStream_t stream) {
  const float* x      = (const float*)d_in[0];
  const float* user_h = (const float*)d_in[1];
  const float* item_h = (const float*)d_in[2];
  const float* W      = (const float*)d_in[3];
  const float* bias   = (const float*)d_in[4];
  const float* A      = (const float*)d_in[5];
  const float* B      = (const float*)d_in[6];
  const float* Wu     = (const float*)d_in[7];
  const float* VU     = (const float*)d_in[8];
  const float* Wi     = (const float*)d_in[9];
  const float* VI     = (const float*)d_in[10];
  float* out = (float*)d_out;

  float* ws   = (float*)d_ws;
  float* Weff = ws;                 // 768*768
  float* wqu  = Weff + D * D;       // 64*768
  float* wqi  = wqu + BS * D;       // 64*768

  // 1. fold LoRA into base weight
  weff_kernel<<<(D * D + 255) / 256, 256, 0, stream>>>(W, A, B, Weff);
  // 2. per-sample wq for both branches
  wq_kernel<<<dim3(D / 256, BS), 256, 0, stream>>>(user_h, Wu, wqu);
  wq_kernel<<<dim3(D / 256, BS), 256, 0, stream>>>(item_h, Wi, wqi);
  // 3. bias + both Kronecker branches -> d_out (full overwrite)
  kron_bias_kernel<<<dim3(SEQ, BS), 256, 0, stream>>>(x, bias, VU, VI, wqu, wqi, out);
  // 4. main WMMA GEMM accumulates on top
  gemm_wmma_kernel<<<dim3(D / 64, (BS * SEQ) / 128), 256, 0, stream>>>(x, Weff, out);
}